// EmittanceModule_9560597201487
// MI455X (gfx1250) — compile-verified
//
#include <hip/hip_runtime.h>
#include <math.h>

typedef __attribute__((ext_vector_type(2))) float v2f;
typedef __attribute__((ext_vector_type(8))) float v8f;

#define WAVES_PER_BLOCK 8
#define D_TUNE 8
#define NSTEP 16
#define HID 256

// Hardware tanh (gfx1250 TRANS op). Builtin if available, else inline asm.
__device__ __forceinline__ float fast_tanh(float v) {
#if defined(__has_builtin)
#if __has_builtin(__builtin_amdgcn_tanhf)
  return __builtin_amdgcn_tanhf(v);
#elif __has_builtin(__builtin_amdgcn_tanh_f32)
  return __builtin_amdgcn_tanh_f32(v);
#else
  float r;
  // v_nop covers the TRANS result-use hazard (ISA 7.4), since the compiler
  // cannot see through the asm to schedule around it.
  asm("v_tanh_f32 %0, %1\n\tv_nop" : "=v"(r) : "v"(v));
  return r;
#endif
#else
  float r;
  asm("v_tanh_f32 %0, %1\n\tv_nop" : "=v"(r) : "v"(v));
  return r;
#endif
}

// ---------------------------------------------------------------------------
// Kernel 1: compute shared pinv(A) (3x16) into d_ws, zero the output scalar.
// A[j] = [k_j^2, k_j, 1];  pinv = (A^T A)^-1 A^T  (full column rank).
// ---------------------------------------------------------------------------
__global__ void emit_prep_kernel(const float* __restrict__ Xm,
                                 float* __restrict__ P,
                                 float* __restrict__ out) {
  if (threadIdx.x == 0) {
    double S1 = 0.0, S2 = 0.0, S3 = 0.0, S4 = 0.0;
    double k[NSTEP];
    for (int j = 0; j < NSTEP; ++j) {
      double kk = (double)Xm[j];
      k[j] = kk;
      double k2 = kk * kk;
      S1 += kk; S2 += k2; S3 += k2 * kk; S4 += k2 * k2;
    }
    // Normal matrix N = [[S4,S3,S2],[S3,S2,S1],[S2,S1,16]] (symmetric)
    double N00 = S4, N01 = S3, N02 = S2, N11 = S2, N12 = S1, N22 = (double)NSTEP;
    double det = N00 * (N11 * N22 - N12 * N12)
               - N01 * (N01 * N22 - N12 * N02)
               + N02 * (N01 * N12 - N11 * N02);
    double inv = 1.0 / det;
    double i00 = (N11 * N22 - N12 * N12) * inv;
    double i01 = (N02 * N12 - N01 * N22) * inv;
    double i02 = (N01 * N12 - N02 * N11) * inv;
    double i11 = (N00 * N22 - N02 * N02) * inv;
    double i12 = (N02 * N01 - N00 * N12) * inv;
    double i22 = (N00 * N11 - N01 * N01) * inv;
    for (int j = 0; j < NSTEP; ++j) {
      double kj = k[j], kj2 = kj * kj;
      P[0 * NSTEP + j] = (float)(i00 * kj2 + i01 * kj + i02);
      P[1 * NSTEP + j] = (float)(i01 * kj2 + i11 * kj + i12);
      P[2 * NSTEP + j] = (float)(i02 * kj2 + i12 * kj + i22);
    }
    out[0] = 0.0f;
  }
}

// ---------------------------------------------------------------------------
// Kernel 2: one wave32 per sample.
//   h = tanh(xs @ W1[s] + b1[s]);  ys = h @ w2[s];  Xc = pinv @ ys;
//   sigs = M @ Xc;  emit2 = s11*s22 - s12^2;  atomic-sum.
// GEMM via V_WMMA_F32_16X16X4_F32, two k-steps (K=0..7 from x[s]); the K=8
// row (X_meas, rank-1) is folded into the accumulator initialization.
// ---------------------------------------------------------------------------
__global__ __launch_bounds__(WAVES_PER_BLOCK * 32)
void emit_wmma_kernel(const float* __restrict__ x,
                      const float* __restrict__ Xm,
                      const float* __restrict__ W1,
                      const float* __restrict__ b1,
                      const float* __restrict__ w2,
                      const float* __restrict__ P,
                      float* __restrict__ out) {
  __shared__ float red[WAVES_PER_BLOCK];

  const int lane = threadIdx.x & 31;
  const int wave = threadIdx.x >> 5;
  const int s    = blockIdx.x * WAVES_PER_BLOCK + wave;
  const int half = lane >> 4;   // 0: K rows {0,1}/{4,5}; 1: K rows {2,3}/{6,7}
  const int hl   = lane & 15;

  const float* __restrict__ xs  = x  + (size_t)s * D_TUNE;
  const float* __restrict__ W1s = W1 + (size_t)s * (9 * HID);
  const float* __restrict__ b1s = b1 + (size_t)s * HID;
  const float* __restrict__ w2s = w2 + (size_t)s * HID;

  // A tiles (16x4 each), ISA layout: VGPR0 = {K0 | K2}, VGPR1 = {K1 | K3}
  const int klo = 2 * half;
  v2f a0, a1;
  a0.x = xs[klo];     a0.y = xs[klo + 1];
  a1.x = xs[4 + klo]; a1.y = xs[5 + klo];

  // X_meas per C-row: C layout has lane -> (n = hl, m = v + 8*half)
  float xmv[8];
#pragma unroll
  for (int v = 0; v < 8; ++v) xmv[v] = Xm[v + 8 * half];

  float ysacc[8];
#pragma unroll
  for (int v = 0; v < 8; ++v) ysacc[v] = 0.0f;

  const int r0 = 2 * half;  // B row base for this half

#pragma unroll 4
  for (int nt = 0; nt < NSTEP; ++nt) {
    const int col = nt * 16 + hl;

    // B tiles (4x16), streamed non-temporally (W1 is 300MB, single use)
    v2f b0v, b1v;
    b0v.x = __builtin_nontemporal_load(&W1s[(r0 + 0) * HID + col]);
    b0v.y = __builtin_nontemporal_load(&W1s[(r0 + 1) * HID + col]);
    b1v.x = __builtin_nontemporal_load(&W1s[(4 + r0) * HID + col]);
    b1v.y = __builtin_nontemporal_load(&W1s[(5 + r0) * HID + col]);
    const float w8 = __builtin_nontemporal_load(&W1s[8 * HID + col]);

    // acc[v] = bias[n] + X_meas[m] * W1[8][n]   (rank-1 K=8 term)
    const float bias = b1s[col];
    v8f acc;
#pragma unroll
    for (int v = 0; v < 8; ++v) acc[v] = fmaf(xmv[v], w8, bias);

    // D = A*B + C  (fp32 matrix pipe)
    acc = __builtin_amdgcn_wmma_f32_16x16x4_f32(false, a0, false, b0v,
                                                (short)0, acc, false, false);
    acc = __builtin_amdgcn_wmma_f32_16x16x4_f32(false, a1, false, b1v,
                                                (short)0, acc, false, false);

    // h = tanh(acc); ys[m] += h[m][n] * w2[n]   (hardware v_tanh_f32)
    const float w2v = w2s[col];
#pragma unroll
    for (int v = 0; v < 8; ++v) ysacc[v] += fast_tanh(acc[v]) * w2v;
  }

  // Reduce over n (16 lanes within each half-wave)
#pragma unroll
  for (int v = 0; v < 8; ++v) {
    float t = ysacc[v];
    t += __shfl_xor(t, 1, 32);
    t += __shfl_xor(t, 2, 32);
    t += __shfl_xor(t, 4, 32);
    t += __shfl_xor(t, 8, 32);
    ysacc[v] = t;
  }

  // Gather full ys[0..15] to all lanes: lanes 0-15 hold m=v, lanes 16-31 m=v+8
  float yall[NSTEP];
#pragma unroll
  for (int v = 0; v < 8; ++v) {
    yall[v]     = __shfl(ysacc[v], 0, 32);
    yall[8 + v] = __shfl(ysacc[v], 16, 32);
  }

  // Xc = pinv @ ys  (uniform scalar work, P via scalar loads)
  float c0 = 0.0f, c1 = 0.0f, c2 = 0.0f;
#pragma unroll
  for (int j = 0; j < NSTEP; ++j) {
    const float y = yall[j];
    c0 = fmaf(P[j], y, c0);
    c1 = fmaf(P[NSTEP + j], y, c1);
    c2 = fmaf(P[2 * NSTEP + j], y, c2);
  }

  // M with D=1, L=0.1: 1/(DL)^2=100, 1/(2DL)=5, 1/(D^3 L)=10, 1/D^2=1
  const float s11 = 100.0f * c0;
  const float s12 = -100.0f * c0 + 5.0f * c1;
  const float s22 = 100.0f * c0 - 10.0f * c1 + c2;
  const float emit2 = s11 * s22 - s12 * s12;

  if (lane == 0) red[wave] = emit2;
  __syncthreads();
  if (threadIdx.x == 0) {
    float sum = 0.0f;
#pragma unroll
    for (int w = 0; w < WAVES_PER_BLOCK; ++w) sum += red[w];
    atomicAdd(out, sum);
  }
}

// ---------------------------------------------------------------------------
extern "C" void kernel_launch(void* const* d_in, const int* in_sizes, int n_in,
                              void* d_out, int out_size, void* d_ws, size_t ws_size,
                              hipStream_t stream) {
  const float* x   = (const float*)d_in[0];  // (S, 8)
  const float* Xm  = (const float*)d_in[1];  // (16,)
  const float* W1  = (const float*)d_in[2];  // (S, 9, 256)
  const float* b1  = (const float*)d_in[3];  // (S, 256)
  const float* w2  = (const float*)d_in[4];  // (S, 256)
  float* out = (float*)d_out;
  float* P   = (float*)d_ws;                 // 3x16 pinv matrix

  const int S = in_sizes[0] / D_TUNE;        // 32768

  emit_prep_kernel<<<1, 32, 0, stream>>>(Xm, P, out);
  emit_wmma_kernel<<<S / WAVES_PER_BLOCK, WAVES_PER_BLOCK * 32, 0, stream>>>(
      x, Xm, W1, b1, w2, P, out);
}